// MultiViewMatchModule_91147795956022
// MI455X (gfx1250) — compile-verified
//
#include <hip/hip_runtime.h>

#define NV        10            // N*V
#define C_DIM     515
#define H_DIM     128
#define W_DIM     128
#define HW        (H_DIM * W_DIM)
#define P_DIM     10
#define CENTER_CH 514
#define VAL_TH    0.3f
#define NEG_INF   (-__builtin_inff())

typedef float v2f __attribute__((ext_vector_type(2)));
typedef float v8f __attribute__((ext_vector_type(8)));

// ---------------------------------------------------------------------------
// Kernel 1: 5x5 NMS (separable max in LDS) + iterative top-10 per (n,v).
// One workgroup per heatmap; both 64KB buffers fit in CDNA5's 320KB LDS.
// ---------------------------------------------------------------------------
__global__ void nms_topk_kernel(const float* __restrict__ fm,
                                float* __restrict__ out_valk)
{
    __shared__ float s_row[HW];     // row-wise 5-max
    __shared__ float s_nms[HW];     // nms-suppressed heatmap
    __shared__ float s_rv[256];
    __shared__ int   s_ri[256];

    const int b   = blockIdx.x;
    const int tid = threadIdx.x;
    const float* hm = fm + ((size_t)b * C_DIM + CENTER_CH) * HW;

    // Pass A: horizontal 5-max (SAME padding == skip OOB)
    for (int i = 0; i < HW / 256; ++i) {
        int idx = i * 256 + tid;
        int x   = idx & (W_DIM - 1);
        float m = NEG_INF;
        #pragma unroll
        for (int dx = -2; dx <= 2; ++dx) {
            int xx = x + dx;
            if (xx >= 0 && xx < W_DIM) m = fmaxf(m, hm[idx - x + xx]);
        }
        s_row[idx] = m;
    }
    __syncthreads();

    // Pass B: vertical 5-max + keep-equal mask
    for (int i = 0; i < HW / 256; ++i) {
        int idx = i * 256 + tid;
        int y   = idx >> 7;
        float m = NEG_INF;
        #pragma unroll
        for (int dy = -2; dy <= 2; ++dy) {
            int yy = y + dy;
            if (yy >= 0 && yy < H_DIM) m = fmaxf(m, s_row[idx + (yy - y) * W_DIM]);
        }
        float h = hm[idx];
        s_nms[idx] = (m == h) ? h : 0.0f;
    }
    __syncthreads();

    // Pass C: top-10 by repeated argmax (tie-break: lowest index, like lax.top_k)
    for (int k = 0; k < P_DIM; ++k) {
        float bv = NEG_INF; int bi = HW;
        for (int i = 0; i < HW / 256; ++i) {
            int idx = i * 256 + tid;
            float v = s_nms[idx];
            if (v > bv || (v == bv && idx < bi)) { bv = v; bi = idx; }
        }
        s_rv[tid] = bv; s_ri[tid] = bi;
        __syncthreads();
        for (int s = 128; s > 0; s >>= 1) {
            if (tid < s) {
                float v2 = s_rv[tid + s]; int i2 = s_ri[tid + s];
                if (v2 > s_rv[tid] || (v2 == s_rv[tid] && i2 < s_ri[tid])) {
                    s_rv[tid] = v2; s_ri[tid] = i2;
                }
            }
            __syncthreads();
        }
        if (tid == 0) {
            float v = s_rv[0];
            out_valk[b * P_DIM + k] = (v < VAL_TH) ? 0.0f : v;
            s_nms[s_ri[0]] = NEG_INF;          // suppress for next round
        }
        __syncthreads();
    }
}

// ---------------------------------------------------------------------------
// Kernel 2: bilinear grid-sample as WMMA f32 GEMM.
// Per batch b: node[p,c] = sum_i w[p,i] * G[4p+i, c]  ==  [16x40]x[40x515]
// with block-diagonal A, done as 10 chained V_WMMA_F32_16X16X4_F32 per
// 16-channel tile. VGPR layouts per ISA 7.12.2:
//   A 16x4 f32: VGPR0 = {K0 | K2}, VGPR1 = {K1 | K3}, M = lane%16
//   B 4x16 f32: VGPR0 = {K0 | K2}, VGPR1 = {K1 | K3}, N = lane%16
//   D 16x16 f32: VGPR r = row r (lanes 0-15) / row 8+r (lanes 16-31)
// ---------------------------------------------------------------------------
__global__ void gridsample_wmma_kernel(const float* __restrict__ fm,
                                       const float* __restrict__ centers,
                                       float* __restrict__ out_nodes)
{
    __shared__ int   s_off[4][P_DIM];   // corner linear offsets (y*W+x)
    __shared__ float s_w[4][P_DIM];     // bilinear weights

    const int b   = blockIdx.x;
    const int tid = threadIdx.x;

    if (tid < P_DIM) {
        const float* mc = centers + (size_t)(b * P_DIM + tid) * 3;
        float x = mc[0], y = mc[1];
        float x0f = floorf(x), y0f = floorf(y);
        float wx = x - x0f, wy = y - y0f;
        int x0 = (int)x0f; x0 = x0 < 0 ? 0 : (x0 > W_DIM - 1 ? W_DIM - 1 : x0);
        int x1 = x0 + 1 > W_DIM - 1 ? W_DIM - 1 : x0 + 1;
        int y0 = (int)y0f; y0 = y0 < 0 ? 0 : (y0 > H_DIM - 1 ? H_DIM - 1 : y0);
        int y1 = y0 + 1 > H_DIM - 1 ? H_DIM - 1 : y0 + 1;
        s_off[0][tid] = y0 * W_DIM + x0;  s_w[0][tid] = (1.f - wx) * (1.f - wy);
        s_off[1][tid] = y0 * W_DIM + x1;  s_w[1][tid] = wx * (1.f - wy);
        s_off[2][tid] = y1 * W_DIM + x0;  s_w[2][tid] = (1.f - wx) * wy;
        s_off[3][tid] = y1 * W_DIM + x1;  s_w[3][tid] = wx * wy;
    }
    __syncthreads();

    const int wave = tid >> 5;
    const int lane = tid & 31;
    const int n    = lane & 15;     // B column / A row owned by this lane
    const int hi   = lane >> 4;     // 0 -> K rows {0,1}, 1 -> K rows {2,3}
    const int NT   = (C_DIM + 15) / 16;   // 33 channel tiles

    for (int t = wave; t < NT; t += 4) {
        int c  = t * 16 + n;
        int cc = c < C_DIM ? c : C_DIM - 1;   // clamp loads; store is guarded
        const float* base = fm + ((size_t)b * C_DIM + cc) * HW;

        v8f acc = {};
        for (int k = 0; k < P_DIM; ++k) {
            // B chunk: gathered corner rows 4k+{hi*2, hi*2+1} at channel c
            v2f bv;
            bv.x = base[s_off[hi * 2 + 0][k]];
            bv.y = base[s_off[hi * 2 + 1][k]];
            // A chunk: only row m==k nonzero, holding bilinear weights
            float act = (n == k) ? 1.0f : 0.0f;
            v2f av;
            av.x = act * s_w[hi * 2 + 0][k];
            av.y = act * s_w[hi * 2 + 1][k];
            acc = __builtin_amdgcn_wmma_f32_16x16x4_f32(
                false, av, false, bv, (short)0, acc, false, false);
        }
        if (c < C_DIM) {
            if (hi == 0) {
                #pragma unroll
                for (int r = 0; r < 8; ++r)
                    out_nodes[(size_t)(b * P_DIM + r) * C_DIM + c] = acc[r];
            } else {
                out_nodes[(size_t)(b * P_DIM + 8) * C_DIM + c] = acc[0];
                out_nodes[(size_t)(b * P_DIM + 9) * C_DIM + c] = acc[1];
            }
        }
    }
}

// ---------------------------------------------------------------------------
// Kernel 3: weighted BCE over valid edges (E=2000 -> single workgroup).
// ---------------------------------------------------------------------------
__global__ void edge_loss_kernel(const float* __restrict__ logits,
                                 const float* __restrict__ labels,
                                 const float* __restrict__ valid,
                                 float* __restrict__ out_loss, int E)
{
    __shared__ float s0[256];
    __shared__ float s1[256];
    const int tid = threadIdx.x;

    float pos = 0.f, tot = 0.f;
    for (int i = tid; i < E; i += 256) {
        pos += labels[i] * valid[i];
        tot += valid[i];
    }
    s0[tid] = pos; s1[tid] = tot;
    __syncthreads();
    for (int s = 128; s > 0; s >>= 1) {
        if (tid < s) { s0[tid] += s0[tid + s]; s1[tid] += s1[tid + s]; }
        __syncthreads();
    }
    const float num_pos = s0[0];
    const float num_neg = s1[0] - num_pos;
    __syncthreads();

    const float wpos = 1.0f / (num_pos + 1e-12f);
    const float wneg = 1.0f / (num_neg + 1e-12f);
    float acc = 0.f;
    for (int i = tid; i < E; i += 256) {
        float x = logits[i], lab = labels[i], v = valid[i];
        float sp     = log1pf(expf(-fabsf(x)));   // stable softplus(-|x|)
        float logp   = fminf(x, 0.f) - sp;        // log_sigmoid(x)
        float log1mp = fminf(-x, 0.f) - sp;       // log_sigmoid(-x)
        float bce    = -(lab * logp + (1.f - lab) * log1mp);
        float w      = (lab > 0.f) ? wpos : wneg;
        acc += bce * w * v;
    }
    s0[tid] = acc;
    __syncthreads();
    for (int s = 128; s > 0; s >>= 1) {
        if (tid < s) s0[tid] += s0[tid + s];
        __syncthreads();
    }
    if (tid == 0) *out_loss = s0[0];
}

// ---------------------------------------------------------------------------
extern "C" void kernel_launch(void* const* d_in, const int* in_sizes, int n_in,
                              void* d_out, int out_size, void* d_ws, size_t ws_size,
                              hipStream_t stream)
{
    const float* fm      = (const float*)d_in[0];   // [2,5,515,128,128]
    const float* centers = (const float*)d_in[1];   // [2,50,3]
    const float* logits  = (const float*)d_in[2];   // [2000]
    const float* labels  = (const float*)d_in[3];
    const float* valid   = (const float*)d_in[4];
    const int E = in_sizes[2];

    float* out       = (float*)d_out;
    float* out_valk  = out;                                   // 100
    float* out_nodes = out + NV * P_DIM;                      // 51500
    float* out_loss  = out + NV * P_DIM + NV * P_DIM * C_DIM; // 1

    nms_topk_kernel<<<NV, 256, 0, stream>>>(fm, out_valk);
    gridsample_wmma_kernel<<<NV, 128, 0, stream>>>(fm, centers, out_nodes);
    edge_loss_kernel<<<1, 256, 0, stream>>>(logits, labels, valid, out_loss, E);
}